// jslab_21878563406226
// MI455X (gfx1250) — compile-verified
//
#include <hip/hip_runtime.h>

typedef __attribute__((ext_vector_type(2))) float v2f;
typedef __attribute__((ext_vector_type(8))) float v8f;

#define NF     720   // forcing intervals / output length
#define NSUB   60    // substeps per interval
#define NCHUNK 90    // chunks of 8 complex intervals (90*8 = 720 >= 719)
#define NTILE  6     // 6 WMMA tiles of 16 chunks (96 columns, padded)
#define CPAD   96

__global__ __launch_bounds__(256)
void slab_linear_scan_wmma(const float* __restrict__ pk,
                           const float* __restrict__ TAx,
                           const float* __restrict__ TAy,
                           const float* __restrict__ fcp,
                           float* __restrict__ out)
{
    __shared__ float sT[16 * 16];        // lower-triangular Toeplitz of A^k (2x2 real blocks)
    __shared__ float sG[CPAD * 16];      // chunked forcing vectors, column-major per chunk
    __shared__ float sS[NCHUNK * 16];    // local scans (WMMA output)
    __shared__ float sP[2 * 9];          // P[k] = A^k, k = 0..8 (complex)
    __shared__ float sCarry[2 * NCHUNK]; // chunk-entry state (complex)
    __shared__ float sC[4];              // c0, c1 (complex interval-forcing constants)

    const int tid = threadIdx.x;

    // ---- Phase 0: scalar constants (single thread; 60-step loops are the only serial core)
    if (tid == 0) {
        float K0 = __expf(pk[0]);
        float K1 = __expf(pk[1]);
        float fc = fcp[0];
        const float dt = 60.0f;
        float ar = 1.0f - dt * K1;   // a = (1 - dt*K1) - i*dt*fc
        float ai = -dt * fc;
        float b  = dt * K0;
        // c0/c1: exact Euler accumulation of interpolation weights through one interval
        float c0r = 0.f, c0i = 0.f, c1r = 0.f, c1i = 0.f;
        for (int n = 0; n < NSUB; n++) {
            float aa  = (float)n / (float)NSUB;
            float t0r = ar * c0r - ai * c0i + b * (1.0f - aa);
            float t0i = ar * c0i + ai * c0r;
            float t1r = ar * c1r - ai * c1i + b * aa;
            float t1i = ar * c1i + ai * c1r;
            c0r = t0r; c0i = t0i; c1r = t1r; c1i = t1i;
        }
        sC[0] = c0r; sC[1] = c0i; sC[2] = c1r; sC[3] = c1i;
        // A = a^60, then P[k] = A^k for k=0..8
        float Ar = 1.0f, Ai = 0.0f;
        for (int n = 0; n < NSUB; n++) {
            float tr = Ar * ar - Ai * ai;
            float ti = Ar * ai + Ai * ar;
            Ar = tr; Ai = ti;
        }
        float pr = 1.0f, pi = 0.0f;
        for (int k = 0; k <= 8; k++) {
            sP[2 * k] = pr; sP[2 * k + 1] = pi;
            float tr = pr * Ar - pi * Ai;
            float ti = pr * Ai + pi * Ar;
            pr = tr; pi = ti;
        }
    }
    // zero the padded chunk columns (90..95)
    for (int i = tid; i < (CPAD - NCHUNK) * 16; i += 256) sG[NCHUNK * 16 + i] = 0.0f;
    __syncthreads();

    // ---- Phase 1: per-interval forcing G[j] = c0*T[j] + c1*T[j+1]  (j = 0..718; pad 719)
    {
        float c0r = sC[0], c0i = sC[1], c1r = sC[2], c1i = sC[3];
        for (int j = tid; j < NCHUNK * 8; j += 256) {
            float gr = 0.f, gi = 0.f;
            if (j < NF - 1) {
                float t0x = TAx[j], t0y = TAy[j];
                float t1x = TAx[j + 1], t1y = TAy[j + 1];
                gr = c0r * t0x - c0i * t0y + c1r * t1x - c1i * t1y;
                gi = c0r * t0y + c0i * t0x + c1r * t1y + c1i * t1x;
            }
            sG[2 * j]     = gr;  // column-major per chunk: offset 16*(j>>3) + 2*(j&7)
            sG[2 * j + 1] = gi;
        }
    }
    // Build T (16x16): block (p,q) = 2x2 real matrix of A^(p-q) for p>=q, else 0.
    {
        int r = tid >> 4, c = tid & 15;
        int p = r >> 1, q = c >> 1, s = r & 1, t = c & 1;
        float v = 0.0f;
        if (p >= q) {
            float mr = sP[2 * (p - q)], mi = sP[2 * (p - q) + 1];
            v = (s == 0) ? ((t == 0) ? mr : -mi)
                         : ((t == 0) ? mi :  mr);
        }
        sT[r * 16 + c] = v;
    }
    __syncthreads();

    // ---- Phase 2: local scans via WMMA.  D(16x16) = T(16x16) x Gtile(16x16)
    //      as 4 accumulating V_WMMA_F32_16X16X4_F32 (K-slices of 4).
    {
        int wave = tid >> 5;
        int lane = tid & 31;
        if (wave < NTILE) {                 // uniform branch: EXEC stays all-ones
            int half = lane >> 4;           // 0 = lanes 0-15, 1 = lanes 16-31
            int l16  = lane & 15;
            int col  = wave * 16 + l16;     // chunk index for B/D columns
            v8f acc = {};
            for (int kb = 0; kb < 4; kb++) {
                v2f afrag;                  // A 16x4: VGPR g, half h -> K = 4*kb + g + 2*h
                afrag[0] = sT[l16 * 16 + 4 * kb + 2 * half + 0];
                afrag[1] = sT[l16 * 16 + 4 * kb + 2 * half + 1];
                v2f bfrag;                  // B 4x16: VGPR g, half h -> row K = 4*kb + g + 2*h
                bfrag[0] = sG[col * 16 + 4 * kb + 2 * half + 0];
                bfrag[1] = sG[col * 16 + 4 * kb + 2 * half + 1];
                acc = __builtin_amdgcn_wmma_f32_16x16x4_f32(
                          false, afrag, false, bfrag, (short)0, acc, false, false);
            }
            if (col < NCHUNK) {
                for (int g = 0; g < 8; g++) {
                    int m = g + 8 * half;   // C/D: VGPR g -> row g (+8 for upper half)
                    sS[col * 16 + m] = acc[g];
                }
            }
        }
    }
    __syncthreads();

    // ---- Phase 3: serial carry scan across 90 chunks (single thread, complex fma)
    if (tid == 0) {
        float A8r = sP[16], A8i = sP[17];   // A^8
        float cr = 0.0f, ci = 0.0f;         // W at interval 0 is 0
        for (int c = 0; c < NCHUNK; c++) {
            sCarry[2 * c] = cr; sCarry[2 * c + 1] = ci;
            float sr = sS[16 * c + 14], si = sS[16 * c + 15]; // S[c][7]
            float nr = A8r * cr - A8i * ci + sr;
            float ni = A8r * ci + A8i * cr + si;
            cr = nr; ci = ni;
        }
    }
    __syncthreads();

    // ---- Phase 4: outputs.  W[m] = S[c][r] + A^(r+1) * carry[c],  m-1 = 8c + r
    for (int m = tid; m < NF; m += 256) {
        float ur, vr;
        if (m == 0) {
            ur = 0.0f; vr = 0.0f;
        } else {
            int j = m - 1;
            int c = j >> 3, r = j & 7;
            float sr = sS[16 * c + 2 * r],  si = sS[16 * c + 2 * r + 1];
            float pr = sP[2 * (r + 1)],     pi = sP[2 * (r + 1) + 1];
            float cr = sCarry[2 * c],       ci = sCarry[2 * c + 1];
            ur = sr + pr * cr - pi * ci;
            vr = si + pr * ci + pi * cr;
        }
        out[m]      = ur;   // U
        out[NF + m] = vr;   // V
    }
}

extern "C" void kernel_launch(void* const* d_in, const int* in_sizes, int n_in,
                              void* d_out, int out_size, void* d_ws, size_t ws_size,
                              hipStream_t stream) {
    (void)in_sizes; (void)n_in; (void)d_ws; (void)ws_size; (void)out_size;
    const float* pk  = (const float*)d_in[0];
    const float* TAx = (const float*)d_in[1];
    const float* TAy = (const float*)d_in[2];
    const float* fc  = (const float*)d_in[3];
    float* out = (float*)d_out;
    slab_linear_scan_wmma<<<1, 256, 0, stream>>>(pk, TAx, TAy, fc, out);
}